// SwinTransformerBlock_40321152974906
// MI455X (gfx1250) — compile-verified
//
#include <hip/hip_runtime.h>
#include <hip/hip_bf16.h>

// Swin block fused into 2 kernels, all GEMMs on V_WMMA_F32_16X16X4_F32 (fp32-exact).
// Kernel1: gather(shift+window)+LN1 -> QKV -> attention -> proj -> scatter xo to d_out.
// Kernel2: LN2 -> MLP(gelu) -> residual, in-place on d_out.

#define DIMC   128
#define IMGH   64
#define IMGW   64
#define NHEAD  4
#define HD     32
#define WINSZ  8
#define NTOK   64      // tokens per window
#define SHIFTA 4

// padded LDS strides (pad by 4 words so bank = 4*row + col pattern, 64 banks)
#define LDX  132       // [64 x 128] tiles
#define LDQ  388       // [64 x 384] qkv
#define LDP  68        // [32 x 64] probs per wave
#define LDH  132       // mlp hidden chunk

typedef __attribute__((ext_vector_type(2))) float v2f;
typedef __attribute__((ext_vector_type(8))) float v8f;

__device__ __forceinline__ v8f wmma4(v2f a, v2f b, v8f c) {
  // D = A(16x4,f32) * B(4x16,f32) + C(16x16,f32)
  return __builtin_amdgcn_wmma_f32_16x16x4_f32(false, a, false, b, (short)0, c,
                                               false, false);
}

__global__ void __launch_bounds__(256)
swin_attn_kernel(const float* __restrict__ x,
                 const float* __restrict__ n1g, const float* __restrict__ n1b,
                 const float* __restrict__ qkv_w, const float* __restrict__ qkv_b,
                 const float* __restrict__ proj_w, const float* __restrict__ proj_b,
                 float* __restrict__ xo)
{
  __shared__ float s_x[NTOK * LDX];        // LN'd window; later reused as attn-out
  __shared__ float s_qkv[NTOK * LDQ];      // q|k|v columns 0..383
  __shared__ float s_p[8 * 32 * LDP];      // per-wave softmax probs (32 rows x 64)

  const int tid   = threadIdx.x;
  const int lane  = tid & 31;
  const int wave  = tid >> 5;
  const int lmod  = lane & 15;             // N (or M) index within tile
  const int lhalf = lane >> 4;             // 0/1 half of wave

  const int wi   = blockIdx.x;             // window id 0..2047
  const int bimg = wi >> 6;                // batch
  const int twin = wi & 63;
  const int th   = twin >> 3, tw = twin & 7;

  // ---------------- Phase 0: gather (roll -SHIFT + window partition) + LayerNorm
  {
    const int tok = tid >> 2;              // 0..63
    const int q   = tid & 3;               // 32-channel slice
    const int wr  = tok >> 3, wc = tok & 7;
    const int h0  = ((th << 3) + wr + SHIFTA) & 63;
    const int w0  = ((tw << 3) + wc + SHIFTA) & 63;
    const float* row = x + (((size_t)bimg << 12) + (h0 << 6) + w0) * DIMC + q * 32;
    float v[32];
    float s = 0.f, ss = 0.f;
#pragma unroll
    for (int i = 0; i < 32; ++i) { float t = row[i]; v[i] = t; s += t; ss += t * t; }
    s  += __shfl_xor(s, 1);  s  += __shfl_xor(s, 2);
    ss += __shfl_xor(ss, 1); ss += __shfl_xor(ss, 2);
    const float mu  = s * (1.f / 128.f);
    const float var = ss * (1.f / 128.f) - mu * mu;
    const float rs  = rsqrtf(var + 1e-5f);
    float* dst = s_x + tok * LDX + q * 32;
    const float* g = n1g + q * 32;
    const float* bb = n1b + q * 32;
#pragma unroll
    for (int i = 0; i < 32; ++i) dst[i] = (v[i] - mu) * rs * g[i] + bb[i];
  }
  __syncthreads();

  // ---------------- Phase 1: QKV = ln @ qkv_w + qkv_b   (M=64,N=384,K=128)
  for (int nti = 0; nti < 3; ++nti) {
    const int n0 = (wave + nti * 8) << 4;        // N-tile base (waves stride 8)
    v8f acc[4] = {};
    for (int k0 = 0; k0 < 128; k0 += 4) {
      const float* bp = qkv_w + (size_t)(k0 + 2 * lhalf) * 384 + n0 + lmod;
      v2f bf = {bp[0], bp[384]};
#pragma unroll
      for (int mt = 0; mt < 4; ++mt) {
        const float* ap = s_x + ((mt << 4) + lmod) * LDX + k0 + 2 * lhalf;
        v2f af = {ap[0], ap[1]};
        acc[mt] = wmma4(af, bf, acc[mt]);
      }
    }
    const float bias = qkv_b[n0 + lmod];
#pragma unroll
    for (int mt = 0; mt < 4; ++mt)
#pragma unroll
      for (int r = 0; r < 8; ++r)
        s_qkv[((mt << 4) + r + (lhalf << 3)) * LDQ + n0 + lmod] = acc[mt][r] + bias;
  }
  __syncthreads();

  // ---------------- Phase 2: per-head S = QK^T * scale, softmax (2 waves/head)
  const int head = wave >> 1;
  const int mh   = wave & 1;               // which 32-row half this wave owns
  const int qoff = head << 5;
  const int koff = 128 + (head << 5);
  const int voff = 256 + (head << 5);
  float* s_pw = s_p + wave * (32 * LDP);

  {
    v8f sacc[2][4] = {};
    for (int nt = 0; nt < 4; ++nt) {
      for (int k0 = 0; k0 < HD; k0 += 4) {
        // B[kk][n] = K[n][kk]
        const float* bp = s_qkv + ((nt << 4) + lmod) * LDQ + koff + k0 + 2 * lhalf;
        v2f bf = {bp[0], bp[1]};
#pragma unroll
        for (int mt = 0; mt < 2; ++mt) {
          const float* ap =
              s_qkv + ((mh << 5) + (mt << 4) + lmod) * LDQ + qoff + k0 + 2 * lhalf;
          v2f af = {ap[0], ap[1]};
          sacc[mt][nt] = wmma4(af, bf, sacc[mt][nt]);
        }
      }
    }
    const float scale = 0.17677669529663687f;  // 1/sqrt(32)
#pragma unroll
    for (int mt = 0; mt < 2; ++mt) {
#pragma unroll
      for (int r = 0; r < 8; ++r) {
        float e0 = sacc[mt][0][r] * scale, e1 = sacc[mt][1][r] * scale;
        float e2 = sacc[mt][2][r] * scale, e3 = sacc[mt][3][r] * scale;
        float mx = fmaxf(fmaxf(e0, e1), fmaxf(e2, e3));
        mx = fmaxf(mx, __shfl_xor(mx, 1));
        mx = fmaxf(mx, __shfl_xor(mx, 2));
        mx = fmaxf(mx, __shfl_xor(mx, 4));
        mx = fmaxf(mx, __shfl_xor(mx, 8));   // reduction stays within 16-lane group
        e0 = expf(e0 - mx); e1 = expf(e1 - mx);
        e2 = expf(e2 - mx); e3 = expf(e3 - mx);
        float sum = e0 + e1 + e2 + e3;
        sum += __shfl_xor(sum, 1);
        sum += __shfl_xor(sum, 2);
        sum += __shfl_xor(sum, 4);
        sum += __shfl_xor(sum, 8);
        const float inv = 1.f / sum;
        const int m = (mt << 4) + r + (lhalf << 3);  // local row 0..31
        float* pr = s_pw + m * LDP + lmod;
        pr[0]  = e0 * inv;
        pr[16] = e1 * inv;
        pr[32] = e2 * inv;
        pr[48] = e3 * inv;
      }
    }
  }

  // ---------------- Phase 3: O = P @ V  -> reuse s_x as [64 x 128] attn output
  {
    v8f oacc[2][2] = {};
    for (int ntv = 0; ntv < 2; ++ntv) {
      for (int k0 = 0; k0 < NTOK; k0 += 4) {
        const float* bp =
            s_qkv + (k0 + 2 * lhalf) * LDQ + voff + (ntv << 4) + lmod;
        v2f bf = {bp[0], bp[LDQ]};
#pragma unroll
        for (int mt = 0; mt < 2; ++mt) {
          const float* ap = s_pw + ((mt << 4) + lmod) * LDP + k0 + 2 * lhalf;
          v2f af = {ap[0], ap[1]};
          oacc[mt][ntv] = wmma4(af, bf, oacc[mt][ntv]);
        }
      }
    }
#pragma unroll
    for (int mt = 0; mt < 2; ++mt)
#pragma unroll
      for (int ntv = 0; ntv < 2; ++ntv)
#pragma unroll
        for (int r = 0; r < 8; ++r)
          s_x[((mh << 5) + (mt << 4) + r + (lhalf << 3)) * LDX +
              (head << 5) + (ntv << 4) + lmod] = oacc[mt][ntv][r];
  }
  __syncthreads();

  // ---------------- Phase 4: proj + scatter (window reverse + roll +SHIFT)
  {
    const int n0 = wave << 4;
    v8f pacc[4] = {};
    for (int k0 = 0; k0 < 128; k0 += 4) {
      const float* bp = proj_w + (size_t)(k0 + 2 * lhalf) * 128 + n0 + lmod;
      v2f bf = {bp[0], bp[128]};
#pragma unroll
      for (int mt = 0; mt < 4; ++mt) {
        const float* ap = s_x + ((mt << 4) + lmod) * LDX + k0 + 2 * lhalf;
        v2f af = {ap[0], ap[1]};
        pacc[mt] = wmma4(af, bf, pacc[mt]);
      }
    }
    const float bias = proj_b[n0 + lmod];
#pragma unroll
    for (int mt = 0; mt < 4; ++mt)
#pragma unroll
      for (int r = 0; r < 8; ++r) {
        const int m  = (mt << 4) + r + (lhalf << 3);  // token in window
        const int wr = m >> 3, wc = m & 7;
        const int h0 = ((th << 3) + wr + SHIFTA) & 63;
        const int w0 = ((tw << 3) + wc + SHIFTA) & 63;
        xo[(((size_t)bimg << 12) + (h0 << 6) + w0) * DIMC + n0 + lmod] =
            pacc[mt][r] + bias;
      }
  }
}

__global__ void __launch_bounds__(256)
swin_mlp_kernel(const float* __restrict__ n2g, const float* __restrict__ n2b,
                const float* __restrict__ w1, const float* __restrict__ b1,
                const float* __restrict__ w2, const float* __restrict__ b2,
                float* __restrict__ io)   // in: xo, out: xo + mlp  (in place)
{
  __shared__ float s_a[64 * LDX];   // LN2(xo)
  __shared__ float s_h[64 * LDH];   // gelu hidden chunk (128 cols)

  const int tid   = threadIdx.x;
  const int lane  = tid & 31;
  const int wave  = tid >> 5;
  const int lmod  = lane & 15;
  const int lhalf = lane >> 4;
  const size_t base = (size_t)blockIdx.x * 64;  // first token of this block

  // LN2
  {
    const int tok = tid >> 2;
    const int q   = tid & 3;
    const float* row = io + (base + tok) * DIMC + q * 32;
    float v[32];
    float s = 0.f, ss = 0.f;
#pragma unroll
    for (int i = 0; i < 32; ++i) { float t = row[i]; v[i] = t; s += t; ss += t * t; }
    s  += __shfl_xor(s, 1);  s  += __shfl_xor(s, 2);
    ss += __shfl_xor(ss, 1); ss += __shfl_xor(ss, 2);
    const float mu  = s * (1.f / 128.f);
    const float var = ss * (1.f / 128.f) - mu * mu;
    const float rs  = rsqrtf(var + 1e-5f);
    float* dst = s_a + tok * LDX + q * 32;
    const float* g = n2g + q * 32;
    const float* bb = n2b + q * 32;
#pragma unroll
    for (int i = 0; i < 32; ++i) dst[i] = (v[i] - mu) * rs * g[i] + bb[i];
  }
  __syncthreads();

  const int n0 = wave << 4;            // this wave's output N-tile column
  v8f oacc[4] = {};

  for (int kc = 0; kc < 4; ++kc) {     // hidden dim 512 in chunks of 128
    // GEMM1 chunk: h = gelu(a @ w1[:, kc*128 : kc*128+128] + b1)
    v8f hacc[4] = {};
    for (int k0 = 0; k0 < 128; k0 += 4) {
      const float* bp =
          w1 + (size_t)(k0 + 2 * lhalf) * 512 + (kc << 7) + n0 + lmod;
      v2f bf = {bp[0], bp[512]};
#pragma unroll
      for (int mt = 0; mt < 4; ++mt) {
        const float* ap = s_a + ((mt << 4) + lmod) * LDX + k0 + 2 * lhalf;
        v2f af = {ap[0], ap[1]};
        hacc[mt] = wmma4(af, bf, hacc[mt]);
      }
    }
    const float bias1 = b1[(kc << 7) + n0 + lmod];
#pragma unroll
    for (int mt = 0; mt < 4; ++mt)
#pragma unroll
      for (int r = 0; r < 8; ++r) {
        float val = hacc[mt][r] + bias1;
        val = 0.5f * val * (1.f + erff(val * 0.7071067811865475f)); // exact gelu
        s_h[((mt << 4) + r + (lhalf << 3)) * LDH + n0 + lmod] = val;
      }
    __syncthreads();

    // GEMM2 accumulate: oacc += h @ w2[kc*128 : kc*128+128, :]
    for (int k0 = 0; k0 < 128; k0 += 4) {
      const float* bp =
          w2 + (size_t)((kc << 7) + k0 + 2 * lhalf) * 128 + n0 + lmod;
      v2f bf = {bp[0], bp[128]};
#pragma unroll
      for (int mt = 0; mt < 4; ++mt) {
        const float* ap = s_h + ((mt << 4) + lmod) * LDH + k0 + 2 * lhalf;
        v2f af = {ap[0], ap[1]};
        oacc[mt] = wmma4(af, bf, oacc[mt]);
      }
    }
    __syncthreads();   // before next chunk overwrites s_h
  }

  const float bias2 = b2[n0 + lmod];
#pragma unroll
  for (int mt = 0; mt < 4; ++mt)
#pragma unroll
    for (int r = 0; r < 8; ++r) {
      const int m = (mt << 4) + r + (lhalf << 3);
      const size_t idx = (base + m) * DIMC + n0 + lmod;
      io[idx] = io[idx] + oacc[mt][r] + bias2;   // residual xo + mlp
    }
}

extern "C" void kernel_launch(void* const* d_in, const int* in_sizes, int n_in,
                              void* d_out, int out_size, void* d_ws, size_t ws_size,
                              hipStream_t stream) {
  (void)in_sizes; (void)n_in; (void)d_ws; (void)ws_size; (void)out_size;
  const float* x      = (const float*)d_in[0];
  const float* n1g    = (const float*)d_in[1];
  const float* n1b    = (const float*)d_in[2];
  const float* qkv_w  = (const float*)d_in[3];
  const float* qkv_b  = (const float*)d_in[4];
  const float* proj_w = (const float*)d_in[5];
  const float* proj_b = (const float*)d_in[6];
  const float* n2g    = (const float*)d_in[7];
  const float* n2b    = (const float*)d_in[8];
  const float* mlp_w1 = (const float*)d_in[9];
  const float* mlp_b1 = (const float*)d_in[10];
  const float* mlp_w2 = (const float*)d_in[11];
  const float* mlp_b2 = (const float*)d_in[12];
  float* out = (float*)d_out;

  const int n_windows = 32 * (IMGH / WINSZ) * (IMGW / WINSZ);   // 2048
  const int n_tok_blk = 32 * IMGH * IMGW / 64;                  // 2048

  swin_attn_kernel<<<n_windows, 256, 0, stream>>>(
      x, n1g, n1b, qkv_w, qkv_b, proj_w, proj_b, out);
  swin_mlp_kernel<<<n_tok_blk, 256, 0, stream>>>(
      n2g, n2b, mlp_w1, mlp_b1, mlp_w2, mlp_b2, out);
}